// Intensity_39221641347494
// MI455X (gfx1250) — compile-verified
//
#include <hip/hip_runtime.h>

#define MAX_MZ   30000
#define WSZ      10
#define VOCAB    26
#define IONS     8
#define BATCH    32768

#define THREADS  256
#define BLOCKS   512
#define NB       (BATCH / BLOCKS)     // 64 batch rows per block

__device__ __constant__ float c_H2O = 18.01056f;
__device__ __constant__ float c_NH3 = 17.02655f;

typedef float v2f __attribute__((ext_vector_type(2)));
typedef float v4f __attribute__((ext_vector_type(4)));
typedef int   v4i __attribute__((ext_vector_type(4)));

#if defined(__has_builtin)
#if __has_builtin(__builtin_amdgcn_global_load_async_to_lds_b128)
#define HAVE_ASYNC_LDS 1
#endif
#endif

__global__ __launch_bounds__(THREADS, 2)
void deepnovo_ion_window_kernel(const float* __restrict__ spectrum,
                                const float* __restrict__ pepmass,
                                const float* __restrict__ prefix_mass,
                                const float* __restrict__ masses,
                                const int*   __restrict__ direction_p,
                                float*       __restrict__ out)
{
    __shared__ float s_spec[MAX_MZ];   // 120 000 B — whole spectrum lives in LDS
    __shared__ float s_pep[NB];
    __shared__ float s_pre[NB];
    __shared__ float s_mass[VOCAB];

    const int tid = threadIdx.x;
    const int b0  = blockIdx.x * NB;

    // ---- stage spectrum into LDS (async DMA path on CDNA5) ----
#if defined(HAVE_ASYNC_LDS)
    {
        v4i* gsrc = (v4i*)spectrum;                 // builtin takes generic int4*
        for (int i = tid; i < MAX_MZ / 4; i += THREADS) {
            __builtin_amdgcn_global_load_async_to_lds_b128(
                gsrc + i,
                (v4i*)(&s_spec[i * 4]),
                0, 0);
        }
    }
#else
    {
        const v4f* gsrc = (const v4f*)spectrum;
        v4f* ldst = (v4f*)s_spec;
        for (int i = tid; i < MAX_MZ / 4; i += THREADS)
            ldst[i] = gsrc[i];
    }
#endif

    for (int i = tid; i < NB; i += THREADS) {
        s_pep[i] = pepmass[b0 + i];
        s_pre[i] = prefix_mass[b0 + i];
    }
    if (tid < VOCAB) s_mass[tid] = masses[tid];

    const int direction = *direction_p;   // uniform scalar load

#if defined(HAVE_ASYNC_LDS)
#if __has_builtin(__builtin_amdgcn_s_wait_asynccnt)
    __builtin_amdgcn_s_wait_asynccnt(0);
#else
    asm volatile("s_wait_asynccnt 0" ::: "memory");
#endif
#endif
    __syncthreads();

    // ---- one task per (b, v, ion); 10 contiguous output floats per task ----
    const int tasks = NB * VOCAB * IONS;                  // 13312
    float* outb = out + (size_t)b0 * (VOCAB * IONS * WSZ);

    for (int t = tid; t < tasks; t += THREADS) {
        const int b   = t / (VOCAB * IONS);
        const int r   = t - b * (VOCAB * IONS);
        const int v   = r >> 3;
        const int ion = r & 7;

        const float pep = s_pep[b];
        const float pre = s_pre[b];
        const float m   = s_mass[v];

        float cb, cy;
        if (direction == 0) { cb = pre + m; cy = pep - cb; }
        else                { cy = pre + m; cb = pep - cy; }

        const float base = (ion < 4) ? cb : cy;
        const int   sub  = ion & 3;
        float ionm;
        if      (sub == 0) ionm = base;
        else if (sub == 1) ionm = base - c_H2O;
        else if (sub == 2) ionm = base - c_NH3;
        else               ionm = base * 0.5f;

        const int  idx   = __float2int_rn(ionm * 10.0f) - (WSZ / 2);
        const bool valid = (idx >= 0) && (idx <= MAX_MZ - WSZ);
        const int  safe  = valid ? idx : 0;
        const float scale = (valid && (v >= 3)) ? 1.0f : 0.0f;

        v2f* op = (v2f*)(outb + (size_t)t * WSZ);   // t*40 bytes: always 8B-aligned
#pragma unroll
        for (int p = 0; p < WSZ / 2; ++p) {
            v2f val;
            val.x = s_spec[safe + 2 * p]     * scale;
            val.y = s_spec[safe + 2 * p + 1] * scale;
            __builtin_nontemporal_store(val, op + p);   // stream past near caches
        }
    }
}

extern "C" void kernel_launch(void* const* d_in, const int* in_sizes, int n_in,
                              void* d_out, int out_size, void* d_ws, size_t ws_size,
                              hipStream_t stream)
{
    const float* spectrum    = (const float*)d_in[0];
    const float* pepmass     = (const float*)d_in[1];
    const float* prefix_mass = (const float*)d_in[2];
    const float* masses      = (const float*)d_in[3];
    const int*   direction   = (const int*)d_in[4];
    float*       out         = (float*)d_out;

    deepnovo_ion_window_kernel<<<dim3(BLOCKS), dim3(THREADS), 0, stream>>>(
        spectrum, pepmass, prefix_mass, masses, direction, out);
}